// KAND_SAT_AGG_82429012345154
// MI455X (gfx1250) — compile-verified
//
#include <hip/hip_runtime.h>

typedef __attribute__((ext_vector_type(2))) float v2f;
typedef __attribute__((ext_vector_type(8))) float v8f;
typedef __attribute__((ext_vector_type(4))) int   v4i;

// ---- fuzzy-logic primitives (LTN defaults, product t-norm family) ----
__device__ __forceinline__ float f_or(float a, float b)      { return a + b - a * b; }
__device__ __forceinline__ float f_implies(float a, float b) { return 1.0f - a + a * b; }

// f[obj][cls]
__device__ __forceinline__ float f_same(const float f[3][3]) {
    float v0 = f[0][0] * f[1][0] * f[2][0];
    float v1 = f[0][1] * f[1][1] * f[2][1];
    float v2 = f[0][2] * f[1][2] * f[2][2];
    return f_or(f_or(v0, v1), v2);
}

__device__ __forceinline__ float f_diff(const float f[3][3]) {
    // perms: (0,1,2),(0,2,1),(1,0,2),(1,2,0),(2,1,0),(2,0,1)
    float t0 = f[0][0] * f[1][1] * f[2][2];
    float t1 = f[0][0] * f[1][2] * f[2][1];
    float t2 = f[0][1] * f[1][0] * f[2][2];
    float t3 = f[0][1] * f[1][2] * f[2][0];
    float t4 = f[0][2] * f[1][1] * f[2][0];
    float t5 = f[0][2] * f[1][0] * f[2][1];
    float out = t5;                 // same chain order as reference
    out = f_or(t4, out);
    out = f_or(t3, out);
    out = f_or(t2, out);
    out = f_or(t1, out);
    out = f_or(t0, out);
    return out;
}

#define BLOCK 256
#define FLOATS_PER_SAMPLE 54
#define TILE_V4 3456   // 256 samples * 54 floats / 4 floats-per-float4

#if __has_builtin(__builtin_amdgcn_global_load_async_to_lds_b128)
#define HAVE_ASYNC_LDS 1
#else
#define HAVE_ASYNC_LDS 0
#endif

// Pointer-to-AS(1)/AS(3) v4i types (attribute attaches to the pointee type).
typedef __attribute__((address_space(1))) v4i* gptr_v4i;
typedef __attribute__((address_space(3))) v4i* lptr_v4i;

#if HAVE_ASYNC_LDS
// Direct memory -> LDS DMA, tracked by ASYNCcnt (gfx1250 async path).
__device__ __forceinline__ void stage16(const float4* g, float4* l) {
    __builtin_amdgcn_global_load_async_to_lds_b128(
        (gptr_v4i)(g), (lptr_v4i)(l), /*offset=*/0, /*cpol=*/0);
}
#else
__device__ __forceinline__ void stage16(const float4* g, float4* l) {
    *l = *g;   // fallback: VMEM load + ds_store
}
#endif

__device__ __forceinline__ void wait_async_then_barrier() {
#if HAVE_ASYNC_LDS
#if __has_builtin(__builtin_amdgcn_s_wait_asynccnt)
    __builtin_amdgcn_s_wait_asynccnt(0);
#else
    asm volatile("s_wait_asynccnt 0x0" ::: "memory");
#endif
#endif
    __syncthreads();
}

__device__ __forceinline__ float lane_f32(float v, int lane) {
    return __int_as_float(__builtin_amdgcn_readlane(__float_as_int(v), lane));
}

__global__ __launch_bounds__(BLOCK)
void kand_main(const float* __restrict__ pCs,
               const float* __restrict__ labels,
               float* __restrict__ partial) {
    __shared__ float4 tile[TILE_V4];        // 55,296 B staged tile
    __shared__ float  wsum[BLOCK / 32];

    const int tid = threadIdx.x;
    const int blk = blockIdx.x;

    // ---- coalesced, 16B-aligned stream of this block's 256 samples into LDS ----
    const float4* __restrict__ src = reinterpret_cast<const float4*>(pCs) + (size_t)blk * TILE_V4;
    #pragma unroll
    for (int i = 0; i < 13; ++i)
        stage16(&src[tid + i * BLOCK], &tile[tid + i * BLOCK]);
    if (tid < TILE_V4 - 13 * BLOCK)         // 128 tail chunks
        stage16(&src[tid + 13 * BLOCK], &tile[tid + 13 * BLOCK]);
    wait_async_then_barrier();

    const float* __restrict__ my = reinterpret_cast<const float*>(tile) + tid * FLOATS_PER_SAMPLE;

    // ---- fuzzy formula per sample ----
    float orf[2];
    #pragma unroll
    for (int mod = 0; mod < 2; ++mod) {     // 0 = shape (concepts 0..2), 1 = color (3..5)
        float s[3], d[3];
        #pragma unroll
        for (int fig = 0; fig < 3; ++fig) {
            float f[3][3];
            #pragma unroll
            for (int o = 0; o < 3; ++o)
                #pragma unroll
                for (int c = 0; c < 3; ++c)
                    f[o][c] = my[fig * 18 + o * 6 + mod * 3 + c];
            s[fig] = f_same(f);
            d[fig] = f_diff(f);
        }
        float p0 = (1.0f - s[0]) * (1.0f - d[0]);
        float p1 = (1.0f - s[1]) * (1.0f - d[1]);
        float p2 = (1.0f - s[2]) * (1.0f - d[2]);
        float and_same = s[0] * (s[1] * s[2]);
        float and_pair = p0 * (p1 * p2);
        float and_diff = d[0] * (d[1] * d[2]);
        orf[mod] = f_or(and_same, f_or(and_pair, and_diff));
    }
    float fin = f_or(orf[0], orf[1]);

    const size_t sid = (size_t)blk * BLOCK + tid;
    float lab = labels[sid];                // perfectly coalesced
    float eq  = f_implies(fin, lab) * f_implies(lab, fin);
    float e   = 1.0f - eq;
    float val = e * e;

    // ---- wave32 reduction via V_WMMA_F32_16X16X4_F32 ----
    // A(16x4 f32): VGPR0 = val (lanes 0-15 -> K=0, lanes 16-31 -> K=2), VGPR1 = 0.
    // B(4x16) = all ones (layout-independent). D[m][n] = val[m] + val[m+16] for all n.
    v2f a; a.x = val;  a.y = 0.0f;
    v2f b; b.x = 1.0f; b.y = 1.0f;
    v8f c = {};
    c = __builtin_amdgcn_wmma_f32_16x16x4_f32(
            /*neg_a=*/false, a, /*neg_b=*/false, b,
            /*c_mod=*/(short)0, c, /*reuse_a=*/false, /*reuse_b=*/false);
    float s8 = c[0] + c[1] + c[2] + c[3] + c[4] + c[5] + c[6] + c[7];
    // lane 0 holds sum over M=0..7 (col 0), lane 16 holds sum over M=8..15 (col 0)
    float wtotal = lane_f32(s8, 0) + lane_f32(s8, 16);

    if ((tid & 31) == 0) wsum[tid >> 5] = wtotal;
    __syncthreads();
    if (tid == 0) {
        float t = 0.0f;
        #pragma unroll
        for (int w = 0; w < BLOCK / 32; ++w) t += wsum[w];
        partial[blk] = t;                   // deterministic: no float atomics
    }
}

__global__ __launch_bounds__(BLOCK)
void kand_finalize(const float* __restrict__ partial,
                   float* __restrict__ out,
                   int nblocks, float invB) {
    __shared__ float sh[BLOCK];
    const int tid = threadIdx.x;
    float s = 0.0f;
    for (int i = tid; i < nblocks; i += BLOCK) s += partial[i];
    sh[tid] = s;
    __syncthreads();
    #pragma unroll
    for (int off = BLOCK / 2; off > 0; off >>= 1) {
        if (tid < off) sh[tid] += sh[tid + off];
        __syncthreads();
    }
    if (tid == 0)
        out[0] = sqrtf(sh[0] * invB + 1e-12f);   // sat_loss = 1 - phi5
}

extern "C" void kernel_launch(void* const* d_in, const int* in_sizes, int n_in,
                              void* d_out, int out_size, void* d_ws, size_t ws_size,
                              hipStream_t stream) {
    const float* pCs    = (const float*)d_in[0];
    const float* labels = (const float*)d_in[1];
    float* out     = (float*)d_out;
    float* partial = (float*)d_ws;          // nblocks floats of scratch

    const int B = in_sizes[1];              // 1,048,576 samples
    const int nblocks = B / BLOCK;          // 4096

    kand_main<<<nblocks, BLOCK, 0, stream>>>(pCs, labels, partial);
    kand_finalize<<<1, BLOCK, 0, stream>>>(partial, out, nblocks, 1.0f / (float)B);
}